// Model_3504693314068
// MI455X (gfx1250) — compile-verified
//
#include <hip/hip_runtime.h>
#include <hip/hip_bf16.h>

// Problem constants (match the reference harness)
#define B_DIM 1024
#define L_DIM 50
#define P_DIM 10
#define D_DIM 64
#define V_DIM 100000

#define RT 4                 // row tiles (16 rows each) per wave, A held in VGPRs
#define COLS_PER_GROUP 10    // column tiles (16 cols each) per wave
#define WAVES_PER_BLOCK 8    // 256 threads, all waves share the column range
#define ROWS_PER_BLOCK (WAVES_PER_BLOCK * RT * 16)   // 512
#define COL_GROUPS (V_DIM / (COLS_PER_GROUP * 16))   // 625 (exact)
#define ROW_BLOCKS (B_DIM / ROWS_PER_BLOCK)          // 2   (exact)

typedef __attribute__((ext_vector_type(2))) float v2f;
typedef __attribute__((ext_vector_type(8))) float v8f;

// ---------------------------------------------------------------------------
// Kernel 1: q[b,d] = user_emb[user_ids[b],d] + low + high + high^2*(S+S^2+S^3)
// One 64-thread block per batch row; thread d owns dimension d.
// ---------------------------------------------------------------------------
__global__ __launch_bounds__(64) void build_q_kernel(
    const int* __restrict__ item_seq,     // [B, L]
    const int* __restrict__ user_ids,     // [B]
    const float* __restrict__ item_emb,   // [NUM_ITEMS, D]
    const float* __restrict__ user_emb,   // [NUM_USERS, D]
    float* __restrict__ qout)             // [B, D]
{
    const int b = blockIdx.x;
    const int d = threadIdx.x;
    const int* __restrict__ seq = item_seq + b * L_DIM;

    float s_all = 0.f, s_p = 0.f;
#pragma unroll 5
    for (int l = 0; l < L_DIM; ++l) {
        const int idx = seq[l];
        const float v = item_emb[(size_t)idx * D_DIM + d];
        s_all += v;
        if (l >= L_DIM - P_DIM) s_p += v;
    }

    const float S    = s_all;
    const float high = s_all * (1.0f / (float)L_DIM);
    const float low  = s_p  * (1.0f / (float)P_DIM);
    const float poly = S + S * S + S * S * S;     // S + S^2 + S^3
    const float un   = low + high + high * high * poly;

    const int u = user_ids[b];
    qout[(size_t)b * D_DIM + d] = user_emb[(size_t)u * D_DIM + d] + un;
}

// ---------------------------------------------------------------------------
// Kernel 2: out[B,V] = q @ W2^T + b2 via V_WMMA_F32_16X16X4_F32.
//  - Each wave holds a 64-row A panel (4 tiles x 16 v2f = 128 VGPRs) in
//    registers and sweeps 10 column tiles -> B-read : store traffic is 1:1.
//  - All 8 waves in a block share the column range -> B tiles hit WGP$/L1,
//    cutting L2 B-traffic ~8x.
//  - Item indices and biases for all 10 tiles are preloaded before the loop,
//    removing the serial items->b2->W2 gather chain (two full-latency waits)
//    from the steady-state iteration.
//  - Output stores are non-temporal: the 410 MB write-once stream must not
//    evict W2 (25.6 MB) from the 192 MB L2.
//
// WMMA operand layout (32-bit, wave32):
//   A 16x4: lanes 0-15 M=lane, K pair {0,1}; lanes 16-31 M=lane-16, K {2,3}
//   B 4x16: mirrored (lane&15 = column N)
//   C/D   : VGPR r -> row (r + 8*half), col (lane & 15)
// ---------------------------------------------------------------------------
__global__ __launch_bounds__(256, 1) void wmma_gemm_kernel(
    const float* __restrict__ q,      // [B, 64]
    const float* __restrict__ W2,     // [V, 64]
    const float* __restrict__ b2,     // [V]
    const int* __restrict__ items,    // [V] items_to_predict
    float* __restrict__ out)          // [B, V]
{
    const int lane = threadIdx.x & 31;
    const int wave = threadIdx.x >> 5;
    const int lid  = lane & 15;       // M (A) / N (B,C,D) index within tile
    const int half = lane >> 4;       // selects K pair {0,1} vs {2,3}

    const int row0     = blockIdx.y * ROWS_PER_BLOCK + wave * (RT * 16);
    const int colTile0 = blockIdx.x * COLS_PER_GROUP;

    // ---- Preload all column-tile item indices, then all biases ----
    int itemIdxArr[COLS_PER_GROUP];
#pragma unroll
    for (int cg = 0; cg < COLS_PER_GROUP; ++cg) {
        itemIdxArr[cg] = items[(colTile0 + cg) * 16 + lid];
    }
    float biasArr[COLS_PER_GROUP];
#pragma unroll
    for (int cg = 0; cg < COLS_PER_GROUP; ++cg) {
        biasArr[cg] = b2[itemIdxArr[cg]];
    }

    // ---- Load the A panel once: 4 row tiles, K=64 -> 16 v2f per tile ----
    v2f areg[RT][16];
#pragma unroll
    for (int t = 0; t < RT; ++t) {
        const v2f* __restrict__ Ar =
            (const v2f*)(q + (size_t)(row0 + t * 16 + lid) * D_DIM);
#pragma unroll
        for (int j = 0; j < 16; ++j) areg[t][j] = Ar[2 * j + half];
    }

    // ---- Sweep column tiles, reloading only B ----
    for (int cg = 0; cg < COLS_PER_GROUP; ++cg) {
        const int col0    = (colTile0 + cg) * 16;
        const float bias  = biasArr[cg];

        const v2f* __restrict__ Br =
            (const v2f*)(W2 + (size_t)itemIdxArr[cg] * D_DIM);
        v2f breg[16];
#pragma unroll
        for (int j = 0; j < 16; ++j) breg[j] = Br[2 * j + half];

        // C starts at the bias (same N for all 8 accumulator rows).
        v8f binit;
#pragma unroll
        for (int r = 0; r < 8; ++r) binit[r] = bias;
        v8f acc[RT];
#pragma unroll
        for (int t = 0; t < RT; ++t) acc[t] = binit;

        // K-step outer, row-tile inner: 4 independent WMMA chains interleave,
        // hiding the WMMA->WMMA RAW hazard latency.
#pragma unroll
        for (int j = 0; j < 16; ++j) {
#pragma unroll
            for (int t = 0; t < RT; ++t) {
                acc[t] = __builtin_amdgcn_wmma_f32_16x16x4_f32(
                    /*neg_a=*/false, areg[t][j], /*neg_b=*/false, breg[j],
                    /*c_mod=*/(short)0, acc[t],
                    /*reuse_a=*/false, /*reuse_b=*/false);
            }
        }

        // Non-temporal stores: write-once stream, keep W2 resident in L2.
#pragma unroll
        for (int t = 0; t < RT; ++t) {
            float* __restrict__ outp =
                out + (size_t)(row0 + t * 16 + 8 * half) * V_DIM
                    + (size_t)col0 + lid;
#pragma unroll
            for (int r = 0; r < 8; ++r) {
                __builtin_nontemporal_store(acc[t][r], outp + (size_t)r * V_DIM);
            }
        }
    }
}

// ---------------------------------------------------------------------------
// Launch wrapper
// Inputs (setup_inputs order):
//   0: item_seq [B*L] int32       1: user_ids [B] int32
//   2: items_to_predict [V] int32 3: item_emb_table [V*D] f32
//   4: user_emb_table [NU*D] f32  5: W2_table [V*D] f32
//   6: b2_table [V] f32
// Output: [B, V] f32. Workspace: q buffer [B*D] f32 (256 KB).
// ---------------------------------------------------------------------------
extern "C" void kernel_launch(void* const* d_in, const int* in_sizes, int n_in,
                              void* d_out, int out_size, void* d_ws, size_t ws_size,
                              hipStream_t stream) {
    const int*   item_seq = (const int*)d_in[0];
    const int*   user_ids = (const int*)d_in[1];
    const int*   items    = (const int*)d_in[2];
    const float* item_emb = (const float*)d_in[3];
    const float* user_emb = (const float*)d_in[4];
    const float* W2       = (const float*)d_in[5];
    const float* b2       = (const float*)d_in[6];
    float*       out      = (float*)d_out;
    float*       qbuf     = (float*)d_ws;   // [B_DIM * D_DIM] floats

    build_q_kernel<<<B_DIM, 64, 0, stream>>>(item_seq, user_ids, item_emb,
                                             user_emb, qbuf);

    dim3 grid(COL_GROUPS, ROW_BLOCKS);      // (625, 2)
    wmma_gemm_kernel<<<grid, 256, 0, stream>>>(qbuf, W2, b2, items, out);
}